// FeaStConv_31138512896570
// MI455X (gfx1250) — compile-verified
//
#include <hip/hip_runtime.h>

typedef __attribute__((ext_vector_type(2))) float v2f;
typedef __attribute__((ext_vector_type(8))) float v8f;

#define F_IN   128
#define F_OUT  128
#define HEADS  2
#define WCOLS  (HEADS * F_OUT)   // 256

#define RT 4   // row tiles (16 rows each) per wave  -> 64 rows
#define CT 2   // col tiles (16 cols each) per wave  -> 32 cols

// ---------------- zero workspace (agg + cnt, contiguous) ----------------
__global__ void feast_zero_kernel(float* __restrict__ buf, long long n) {
    long long i = (long long)blockIdx.x * blockDim.x + threadIdx.x;
    if (i < n) buf[i] = 0.0f;
}

// ---------------- p = x @ u   [N, HEADS]  (one wave per node) ----------------
__global__ void feast_attn_proj_kernel(const float* __restrict__ x,
                                       const float* __restrict__ u,
                                       float* __restrict__ p, int n_nodes) {
    int wave = (int)(((long long)blockIdx.x * blockDim.x + threadIdx.x) >> 5);
    int lane = threadIdx.x & 31;
    if (wave >= n_nodes) return;
    const float4* xr = (const float4*)(x + (size_t)wave * F_IN);
    float4 xv = xr[lane];                 // 32 lanes * 4 = 128 features
    int f0 = lane * 4;
    float s0 = xv.x * u[(f0 + 0) * HEADS]     + xv.y * u[(f0 + 1) * HEADS]
             + xv.z * u[(f0 + 2) * HEADS]     + xv.w * u[(f0 + 3) * HEADS];
    float s1 = xv.x * u[(f0 + 0) * HEADS + 1] + xv.y * u[(f0 + 1) * HEADS + 1]
             + xv.z * u[(f0 + 2) * HEADS + 1] + xv.w * u[(f0 + 3) * HEADS + 1];
    #pragma unroll
    for (int off = 16; off > 0; off >>= 1) {
        s0 += __shfl_down(s0, off, 32);
        s1 += __shfl_down(s1, off, 32);
    }
    if (lane == 0) {
        p[(size_t)wave * 2 + 0] = s0;
        p[(size_t)wave * 2 + 1] = s1;
    }
}

// ---------------- xw = x @ w  [N, 256] via V_WMMA_F32_16X16X4_F32 ----------------
// Register-blocked: each wave owns a 64x32 output block (4x2 tiles of 16x16).
// Per k-step: 4 A-frag b64 loads + 4 B-frag b32 loads feed 8 WMMAs (1.0 wmma/load).
__global__ void feast_gemm_wmma_kernel(const float* __restrict__ x,
                                       const float* __restrict__ w,
                                       float* __restrict__ xw, int n_nodes) {
    int wave = (int)(((long long)blockIdx.x * blockDim.x + threadIdx.x) >> 5);
    int lane = threadIdx.x & 31;
    const int col_blocks = WCOLS / (CT * 16);        // 8
    int row_block = wave / col_blocks;
    int col_block = wave - row_block * col_blocks;
    int m0 = row_block * (RT * 16);
    if (m0 >= n_nodes) return;                       // wave-uniform: EXEC stays all-1s

    int laneM = lane & 15;
    int khalf = (lane >> 4) * 2;                     // lanes 0-15 -> K+{0,1}; 16-31 -> K+{2,3}

    const float* xrow[RT];
    #pragma unroll
    for (int r = 0; r < RT; ++r) {
        int m = m0 + r * 16 + laneM;
        if (m >= n_nodes) m = n_nodes - 1;           // clamp (stores are guarded below)
        xrow[r] = x + (size_t)m * F_IN;
    }
    int ncol[CT];
    #pragma unroll
    for (int t = 0; t < CT; ++t)
        ncol[t] = col_block * (CT * 16) + t * 16 + laneM;

    v8f acc[RT][CT];
    #pragma unroll
    for (int r = 0; r < RT; ++r)
        #pragma unroll
        for (int t = 0; t < CT; ++t)
            acc[r][t] = (v8f){};

    #pragma unroll
    for (int k = 0; k < F_IN; k += 4) {
        v2f a[RT], b[CT];
        #pragma unroll
        for (int r = 0; r < RT; ++r) {               // A 16x4 frags (ISA 7.12.2)
            a[r].x = xrow[r][k + khalf];
            a[r].y = xrow[r][k + khalf + 1];
        }
        #pragma unroll
        for (int t = 0; t < CT; ++t) {               // B 4x16 frags
            b[t].x = w[(size_t)(k + khalf)     * WCOLS + ncol[t]];
            b[t].y = w[(size_t)(k + khalf + 1) * WCOLS + ncol[t]];
        }
        #pragma unroll
        for (int r = 0; r < RT; ++r)
            #pragma unroll
            for (int t = 0; t < CT; ++t)
                acc[r][t] = __builtin_amdgcn_wmma_f32_16x16x4_f32(
                    false, a[r], false, b[t], (short)0, acc[r][t], false, false);
    }

    // C/D layout: VGPR v -> row v + 8*(lane>=16), col = lane&15
    int moff = (lane >> 4) ? 8 : 0;
    #pragma unroll
    for (int r = 0; r < RT; ++r) {
        #pragma unroll
        for (int t = 0; t < CT; ++t) {
            #pragma unroll
            for (int v = 0; v < 8; ++v) {
                int row = m0 + r * 16 + moff + v;
                if (row < n_nodes)
                    xw[(size_t)row * WCOLS + ncol[t]] = acc[r][t][v];
            }
        }
    }
}

// ---------------- per-edge softmax blend + scatter-add (one wave per edge) ----------------
// Edges [0, n_edges) come from edge_index; edges [n_edges, n_edges+n_nodes) are self-loops.
__global__ void feast_edge_kernel(const long long* __restrict__ ei,
                                  const float* __restrict__ p,
                                  const float* __restrict__ c,
                                  const float* __restrict__ xw,
                                  float* __restrict__ agg,
                                  float* __restrict__ cnt,
                                  int n_edges, int n_nodes) {
    int wave = (int)(((long long)blockIdx.x * blockDim.x + threadIdx.x) >> 5);
    int lane = threadIdx.x & 31;
    long long total = (long long)n_edges + n_nodes;
    if (wave >= total) return;

    int s, d;
    if (wave < n_edges) {
        s = (int)ei[wave];                   // row 0: src (j)
        d = (int)ei[(size_t)n_edges + wave]; // row 1: dst (i)
    } else {
        s = d = wave - n_edges;              // self loop
    }

    // attention logits via precomputed projections: (x_j - x_i) @ u + c
    const float2* p2 = (const float2*)p;
    float2 ps = p2[s];
    float2 pd = p2[d];
    float a0 = ps.x - pd.x + c[0];
    float a1 = ps.y - pd.y + c[1];
    float mx = fmaxf(a0, a1);
    float e0 = __expf(a0 - mx);
    float e1 = __expf(a1 - mx);
    float inv = 1.0f / (e0 + e1);
    float q0 = e0 * inv;
    float q1 = e1 * inv;

    // blend the two heads of xw[src] and scatter-add into agg[dst]
    const float4* h0 = (const float4*)(xw + (size_t)s * WCOLS);
    const float4* h1 = (const float4*)(xw + (size_t)s * WCOLS + F_OUT);
    float4 v0 = h0[lane];
    float4 v1 = h1[lane];
    float4 msg;
    msg.x = q0 * v0.x + q1 * v1.x;
    msg.y = q0 * v0.y + q1 * v1.y;
    msg.z = q0 * v0.z + q1 * v1.z;
    msg.w = q0 * v0.w + q1 * v1.w;

    float* dp = agg + (size_t)d * F_OUT + lane * 4;
    atomicAdd(dp + 0, msg.x);
    atomicAdd(dp + 1, msg.y);
    atomicAdd(dp + 2, msg.z);
    atomicAdd(dp + 3, msg.w);
    if (lane == 0) atomicAdd(cnt + d, 1.0f);
}

// ---------------- out = x + relu(agg / max(cnt,1) + bias) ----------------
__global__ void feast_finalize_kernel(const float* __restrict__ x,
                                      const float* __restrict__ agg,
                                      const float* __restrict__ cnt,
                                      const float* __restrict__ bias,
                                      float* __restrict__ out, int n_nodes) {
    long long idx = (long long)blockIdx.x * blockDim.x + threadIdx.x;
    long long total = (long long)n_nodes * F_OUT;
    if (idx >= total) return;
    int node = (int)(idx >> 7);
    int f    = (int)(idx & (F_OUT - 1));
    float v = agg[idx] / fmaxf(cnt[node], 1.0f) + bias[f];
    out[idx] = x[idx] + fmaxf(v, 0.0f);
}

extern "C" void kernel_launch(void* const* d_in, const int* in_sizes, int n_in,
                              void* d_out, int out_size, void* d_ws, size_t ws_size,
                              hipStream_t stream) {
    const float*     x    = (const float*)d_in[0];
    const long long* ei   = (const long long*)d_in[1];   // int64 edge_index [2, E]
    const float*     u    = (const float*)d_in[2];
    const float*     c    = (const float*)d_in[3];
    const float*     w    = (const float*)d_in[4];
    const float*     bias = (const float*)d_in[5];
    float*           out  = (float*)d_out;

    int n_nodes = in_sizes[0] / F_IN;
    int n_edges = in_sizes[1] / 2;

    // workspace layout: xw | p | agg | cnt  (agg,cnt contiguous for one-shot zeroing)
    float* xw  = (float*)d_ws;
    float* p   = xw  + (size_t)n_nodes * WCOLS;
    float* agg = p   + (size_t)n_nodes * HEADS;
    float* cnt = agg + (size_t)n_nodes * F_OUT;

    const int threads = 256;   // 8 waves per block (wave32)
    const int wpb = threads / 32;

    {   // zero agg + cnt
        long long nz = (long long)n_nodes * F_OUT + n_nodes;
        int blocks = (int)((nz + threads - 1) / threads);
        feast_zero_kernel<<<blocks, threads, 0, stream>>>(agg, nz);
    }
    {   // p = x @ u
        int blocks = (n_nodes + wpb - 1) / wpb;
        feast_attn_proj_kernel<<<blocks, threads, 0, stream>>>(x, u, p, n_nodes);
    }
    {   // xw = x @ w  (register-blocked WMMA)
        int row_blocks = (n_nodes + RT * 16 - 1) / (RT * 16);
        long long waves = (long long)row_blocks * (WCOLS / (CT * 16));
        int blocks = (int)((waves + wpb - 1) / wpb);
        feast_gemm_wmma_kernel<<<blocks, threads, 0, stream>>>(x, w, xw, n_nodes);
    }
    {   // per-edge blend + scatter
        long long total = (long long)n_edges + n_nodes;
        int blocks = (int)((total + wpb - 1) / wpb);
        feast_edge_kernel<<<blocks, threads, 0, stream>>>(ei, p, c, xw, agg, cnt,
                                                          n_edges, n_nodes);
    }
    {   // finalize
        long long total = (long long)n_nodes * F_OUT;
        int blocks = (int)((total + threads - 1) / threads);
        feast_finalize_kernel<<<blocks, threads, 0, stream>>>(x, agg, cnt, bias, out, n_nodes);
    }
}